// ICNN_44341242364660
// MI455X (gfx1250) — compile-verified
//
#include <hip/hip_runtime.h>
#include <math.h>

typedef __attribute__((ext_vector_type(2))) float v2f;
typedef __attribute__((ext_vector_type(8))) float v8f;

#define ACT_SCALE (1.0f / 12.0f)
#define LOG2E 1.4426950408889634f
#define LN2   0.6931471805599453f

// Compiler-only barrier: blocks LICM from hoisting LDS loads, emits no code.
#define OPT_BARRIER() asm volatile("" ::: "memory")

// e = exp(-|v|) via raw v_exp_f32 (2^x); arg <= 0 so no overflow path needed.
__device__ __forceinline__ float exp_negabs(float v) {
    return __builtin_amdgcn_exp2f(fabsf(v) * -LOG2E);
}

// Stable softplus + sigmoid, raw hardware trans ops (inputs are range-safe):
// t = 1+e in (1,2] -> v_log_f32 needs no denorm/inf fixups.
__device__ __forceinline__ void sp_sig(float v, float& sp, float& sg) {
    float e  = exp_negabs(v);
    float t  = 1.0f + e;
    float r  = __builtin_amdgcn_rcpf(t);
    float l2 = __builtin_amdgcn_logf(t);
    sp = fmaf(LN2, l2, fmaxf(v, 0.0f));
    sg = (v >= 0.0f) ? r : e * r;
}

__device__ __forceinline__ float sig_only(float v) {
    float e = exp_negabs(v);
    float r = __builtin_amdgcn_rcpf(1.0f + e);
    return (v >= 0.0f) ? r : e * r;
}

__device__ __forceinline__ float sp_only(float v) {
    float e = exp_negabs(v);
    return fmaf(LN2, __builtin_amdgcn_logf(1.0f + e), fmaxf(v, 0.0f));
}

// ---------------------------------------------------------------------------
// Workspace layout (floats):
//   [0,1024)    sp(raw_A_0) row-major 32x32
//   [1024,2048) sp(raw_A_1)
//   [2048,2080) FLw0  (B-order: slot = h*16 + c*2 + g  -> j = 4c+2h+g)
//   [2080,2112) FLw1
//   [2112,2144) FLb
//   [2144,2336) mid tables, per layer l: MB0,MB1,MBB (D-order: slot = h*16+t*8+p -> j=16t+p+8h)
//   [2336,2368) AO = sp(raw_A_out) in D-order
//   [2368]      C1 = B_out_w[0] - dWdI1_ref   (= -dz1_ref . a_out)
//   [2369]      C2 = B_out_w[1]
// ---------------------------------------------------------------------------
#define WS_FLOATS 2370

__global__ void icnn_prep(
    const float* __restrict__ B_first_w, const float* __restrict__ B_first_b,
    const float* __restrict__ raw_A_0, const float* __restrict__ raw_A_1,
    const float* __restrict__ Bs_w_0, const float* __restrict__ Bs_b_0,
    const float* __restrict__ Bs_w_1, const float* __restrict__ Bs_b_1,
    const float* __restrict__ raw_A_out, const float* __restrict__ B_out_w,
    float* __restrict__ ws)
{
    int tid = threadIdx.x;
    for (int i = tid; i < 1024; i += blockDim.x) {
        ws[i]        = sp_only(raw_A_0[i]);
        ws[1024 + i] = sp_only(raw_A_1[i]);
    }
    if (tid < 32) {
        int i = tid;
        int h = i >> 4, r = i & 15;
        // first layer, B-operand order
        int c = r >> 1, g = r & 1;
        int j = 4 * c + 2 * h + g;
        ws[2048 + i] = B_first_w[2 * j + 0];
        ws[2080 + i] = B_first_w[2 * j + 1];
        ws[2112 + i] = B_first_b[j];
        // mid layers + a_out, D (accumulator) order
        int t = r >> 3, p = r & 7;
        int jd = 16 * t + p + 8 * h;
        ws[2144 + i] = Bs_w_0[2 * jd + 0];
        ws[2176 + i] = Bs_w_0[2 * jd + 1];
        ws[2208 + i] = Bs_b_0[jd];
        ws[2240 + i] = Bs_w_1[2 * jd + 0];
        ws[2272 + i] = Bs_w_1[2 * jd + 1];
        ws[2304 + i] = Bs_b_1[jd];
        ws[2336 + i] = sp_only(raw_A_out[jd]);
    }
    __syncthreads();
    if (tid == 0) {
        // forward-mode gradient at z0 = (0,0): only dz/dI1 needed
        float z[32], d1[32];
        for (int j = 0; j < 32; ++j) {
            float spv, sgv; sp_sig(B_first_b[j], spv, sgv);
            z[j] = spv; d1[j] = sgv * B_first_w[2 * j];
        }
        for (int l = 0; l < 2; ++l) {
            const float* A  = ws + l * 1024;
            const float* Bw = l ? Bs_w_1 : Bs_w_0;
            const float* Bb = l ? Bs_b_1 : Bs_b_0;
            float zn[32], dn[32];
            for (int j = 0; j < 32; ++j) {
                float u = 0.0f, ud = 0.0f;
                for (int k = 0; k < 32; ++k) {
                    u  = fmaf(A[j * 32 + k], z[k],  u);
                    ud = fmaf(A[j * 32 + k], d1[k], ud);
                }
                float spu, sgu; sp_sig(u, spu, sgu);
                float spv, sgv; sp_sig(Bb[j], spv, sgv);
                zn[j] = ACT_SCALE * spu * spu + spv;
                dn[j] = 2.0f * ACT_SCALE * spu * sgu * ud + sgv * Bw[2 * j];
            }
            for (int j = 0; j < 32; ++j) { z[j] = zn[j]; d1[j] = dn[j]; }
        }
        float acc = 0.0f;
        for (int j = 0; j < 32; ++j) acc = fmaf(d1[j], sp_only(raw_A_out[j]), acc);
        ws[2368] = -acc;          // C1 folds B_out_w[0] - dWdI1_ref
        ws[2369] = B_out_w[1];    // C2
    }
}

// ---------------------------------------------------------------------------
// D-layout (lane = sample%16; value j = 16t + p + 8h in reg D[t][p]) to
// B-operand layout (B[c][g] holds k = 4c + 2h + g). Half-wave swap + select.
// ---------------------------------------------------------------------------
__device__ __forceinline__ void relayout_D_to_B(const float D[2][8], v2f B[8], int h) {
    float S[2][8];
#pragma unroll
    for (int t = 0; t < 2; ++t)
#pragma unroll
        for (int p = 0; p < 8; ++p)
            S[t][p] = __shfl_xor(D[t][p], 16, 32);
#pragma unroll
    for (int c = 0; c < 8; ++c) {
        const int t = c >> 2, c4 = c & 3;
#pragma unroll
        for (int g = 0; g < 2; ++g) {
            float cand0, cand1;
            if (c4 < 2) { cand0 = D[t][4 * c4 + g];       cand1 = S[t][4 * c4 + 2 + g]; }
            else        { cand0 = S[t][4 * (c4 - 2) + g]; cand1 = D[t][4 * (c4 - 2) + 2 + g]; }
            B[c][g] = h ? cand1 : cand0;
        }
    }
}

__global__ __launch_bounds__(256) void icnn_main(
    const float* __restrict__ eps, const float* __restrict__ ws,
    float* __restrict__ out, int batch, int ntiles)
{
    __shared__ float lds[WS_FLOATS];
    for (int i = threadIdx.x; i < WS_FLOATS; i += blockDim.x) lds[i] = ws[i];
    __syncthreads();

    const int lane = threadIdx.x & 31;
    const int h    = lane >> 4;      // half-wave: which K/feature half this lane carries
    const int lm   = lane & 15;      // sample index within tile
    const int wavesPerBlock = blockDim.x >> 5;
    const int wave   = blockIdx.x * wavesPerBlock + (threadIdx.x >> 5);
    const int nwaves = gridDim.x * wavesPerBlock;

    const float C1 = lds[2368], C2 = lds[2369];

    const v8f vzero = {0.f, 0.f, 0.f, 0.f, 0.f, 0.f, 0.f, 0.f};

    for (int tile = wave; tile < ntiles; tile += nwaves) {   // wave-uniform: EXEC stays full
        // Keep table reads inside the loop as plain ds_loads (no LICM hoisting
        // into 200+ VGPRs); emits no instructions itself.
        OPT_BARRIER();

        int s  = tile * 16 + lm;
        int sl = s < batch ? s : batch - 1;
        float e11 = eps[3 * sl + 0], e22 = eps[3 * sl + 1], e12 = eps[3 * sl + 2];
        float I1 = e11 + e22;
        float I2 = fmaf(e11, e11, fmaf(2.0f * e12, e12, e22 * e22));

        // ---- first layer (K=2, elementwise) directly in B-operand layout ----
        v2f zB[8], d1B[8], d2B[8];
#pragma unroll
        for (int c = 0; c < 8; ++c)
#pragma unroll
            for (int g = 0; g < 2; ++g) {
                int i = h * 16 + c * 2 + g;
                float w0 = lds[2048 + i], w1 = lds[2080 + i];
                float v = fmaf(w0, I1, fmaf(w1, I2, lds[2112 + i]));
                float spv, sgv; sp_sig(v, spv, sgv);
                zB[c][g] = spv; d1B[c][g] = sgv * w0; d2B[c][g] = sgv * w1;
            }

        // ---- two hidden layers: WMMA GEMMs + elementwise + relayout ----
        float d1D[2][8], d2D[2][8];
#pragma unroll
        for (int l = 0; l < 2; ++l) {
            OPT_BARRIER();   // partition table live-ranges per layer
            v8f az[2], a1[2], a2[2];
#pragma unroll
            for (int t = 0; t < 2; ++t) {
                az[t] = vzero; a1[t] = vzero; a2[t] = vzero;
#pragma unroll
                for (int c = 0; c < 8; ++c) {
                    // A operand: sp(rawA)[16t + lane%16][4c + 2h .. +1]  (even -> 8B aligned ds_load_b64)
                    v2f a = *(const v2f*)(lds + l * 1024 + (16 * t + lm) * 32 + 4 * c + 2 * h);
                    az[t] = __builtin_amdgcn_wmma_f32_16x16x4_f32(false, a, false, zB[c],  (short)0, az[t], false, false);
                    a1[t] = __builtin_amdgcn_wmma_f32_16x16x4_f32(false, a, false, d1B[c], (short)0, a1[t], false, false);
                    a2[t] = __builtin_amdgcn_wmma_f32_16x16x4_f32(false, a, false, d2B[c], (short)0, a2[t], false, false);
                }
            }
            float znD[2][8];
#pragma unroll
            for (int t = 0; t < 2; ++t)
#pragma unroll
                for (int p = 0; p < 8; ++p) {
                    int i = l * 96 + h * 16 + t * 8 + p;
                    float w0 = lds[2144 + i], w1 = lds[2176 + i];
                    float u = az[t][p];
                    float spu, sgu; sp_sig(u, spu, sgu);
                    float v = fmaf(w0, I1, fmaf(w1, I2, lds[2208 + i]));
                    float sgv;
                    if (l == 0) {
                        float spv; sp_sig(v, spv, sgv);
                        znD[t][p] = fmaf(ACT_SCALE * spu, spu, spv);
                    } else {
                        sgv = sig_only(v);   // z_new unused after last layer: skip its log
                    }
                    float gf = (2.0f * ACT_SCALE) * spu * sgu;
                    d1D[t][p] = fmaf(gf, a1[t][p], sgv * w0);
                    d2D[t][p] = fmaf(gf, a2[t][p], sgv * w1);
                }
            if (l == 0) {
                relayout_D_to_B(znD, zB,  h);
                relayout_D_to_B(d1D, d1B, h);
                relayout_D_to_B(d2D, d2B, h);
            }
        }

        // ---- output layer: grad only; dot with a_out, combine halves ----
        float g1 = 0.0f, g2 = 0.0f;
#pragma unroll
        for (int t = 0; t < 2; ++t)
#pragma unroll
            for (int p = 0; p < 8; ++p) {
                float ao = lds[2336 + h * 16 + t * 8 + p];
                g1 = fmaf(d1D[t][p], ao, g1);
                g2 = fmaf(d2D[t][p], ao, g2);
            }
        g1 += __shfl_xor(g1, 16, 32);
        g2 += __shfl_xor(g2, 16, 32);
        float dI1 = g1 + C1, dI2 = g2 + C2;
        if (h == 0 && s < batch) {
            out[3 * s + 0] = fmaf(2.0f * e11, dI2, dI1);
            out[3 * s + 1] = fmaf(2.0f * e22, dI2, dI1);
            out[3 * s + 2] = 2.0f * e12 * dI2;
        }
    }
}

extern "C" void kernel_launch(void* const* d_in, const int* in_sizes, int n_in,
                              void* d_out, int out_size, void* d_ws, size_t ws_size,
                              hipStream_t stream)
{
    const float* eps = (const float*)d_in[0];
    const float* Bfw = (const float*)d_in[1];
    const float* Bfb = (const float*)d_in[2];
    const float* rA0 = (const float*)d_in[3];
    const float* rA1 = (const float*)d_in[4];
    const float* Bw0 = (const float*)d_in[5];
    const float* Bb0 = (const float*)d_in[6];
    const float* Bw1 = (const float*)d_in[7];
    const float* Bb1 = (const float*)d_in[8];
    const float* rAo = (const float*)d_in[9];
    const float* Bow = (const float*)d_in[10];
    // d_in[11] (B_out_b) cancels in the gradient.

    float* ws  = (float*)d_ws;
    float* out = (float*)d_out;
    int batch  = in_sizes[0] / 3;
    int ntiles = (batch + 15) / 16;

    icnn_prep<<<1, 256, 0, stream>>>(Bfw, Bfb, rA0, rA1, Bw0, Bb0, Bw1, Bb1, rAo, Bow, ws);
    icnn_main<<<2048, 256, 0, stream>>>(eps, ws, out, batch, ntiles);
}